// GraphSAGE_15015205667253
// MI455X (gfx1250) — compile-verified
//
#include <hip/hip_runtime.h>

// ---------------------------------------------------------------------------
// GraphSAGE (2-layer SAGEConv, mean aggregation) for MI455X / gfx1250.
// Aggregation: edge-parallel float4 gathers + global f32 atomics (L2-resident:
// whole working set < 192MB L2). GEMMs: V_WMMA_F32_16X16X4_F32, one wave per
// 16x16 output tile, degree-normalization / bias / ReLU fused into operand
// load & epilogue. log_softmax as a node-parallel epilogue kernel.
// ---------------------------------------------------------------------------

typedef __attribute__((ext_vector_type(2))) float v2f;
typedef __attribute__((ext_vector_type(8))) float v8f;

#define NNODES 50000
#define IN_CH  128
#define HID_CH 256
#define OUT_CH 40

// ---------------------------------------------------------------- utilities
__global__ void zero_f32(float* __restrict__ p, int n) {
  int i = blockIdx.x * blockDim.x + threadIdx.x;
  if (i < n) p[i] = 0.0f;
}

__global__ void degree_kernel(const int* __restrict__ dst,
                              float* __restrict__ deg, int E) {
  int e = blockIdx.x * blockDim.x + threadIdx.x;
  if (e < E) atomicAdd(&deg[dst[e]], 1.0f);
}

// One thread per (edge, 4-channel group): float4 gather + 4 f32 atomics.
template <int C>
__global__ void scatter_add(const int* __restrict__ src,
                            const int* __restrict__ dst,
                            const float* __restrict__ feat,
                            float* __restrict__ agg, int E) {
  const int GP = C / 4;  // power of two -> shifts
  int t = blockIdx.x * blockDim.x + threadIdx.x;
  if (t >= E * GP) return;
  int e = t / GP;
  int g = t % GP;
  int s = src[e], d = dst[e];
  const float4 v = *(const float4*)(feat + (long long)s * C + g * 4);
  float* o = agg + (long long)d * C + g * 4;
  atomicAdd(o + 0, v.x);
  atomicAdd(o + 1, v.y);
  atomicAdd(o + 2, v.z);
  atomicAdd(o + 3, v.w);
}

// ------------------------------------------------------------- layer 1 GEMM
// h = relu( (agg/deg) @ W1l.T + x @ W1r.T + b1 )   [50000,128] -> [50000,256]
// Block: 512 threads = 16 waves; blockIdx = 16-node tile; wave = 16-col tile.
__global__ __launch_bounds__(512) void gemm1_wmma(
    const float* __restrict__ x, const float* __restrict__ agg,
    const float* __restrict__ deg, const float* __restrict__ W1l,
    const float* __restrict__ W1r, const float* __restrict__ b1,
    float* __restrict__ h) {
  const int lane = threadIdx.x & 31;
  const int wave = threadIdx.x >> 5;   // N tile: 0..15
  const int hi   = lane >> 4;          // 0: K=0,1  1: K=2,3
  const int m    = lane & 15;
  const int row  = blockIdx.x * 16 + m;        // node (A operand)
  const int col  = wave * 16 + m;              // hidden channel (B operand)
  const float invd = 1.0f / fmaxf(deg[row], 1.0f);

  const float* xr = x   + (long long)row * IN_CH;
  const float* ar = agg + (long long)row * IN_CH;
  const float* wl = W1l + (long long)col * IN_CH;  // B[k][col] = W1l[col][k]
  const float* wr = W1r + (long long)col * IN_CH;

  v8f acc = {};
  for (int kk = 0; kk < IN_CH; kk += 4) {
    const int k0 = kk + 2 * hi;
    v2f aA = {ar[k0] * invd, ar[k0 + 1] * invd};
    v2f aX = {xr[k0], xr[k0 + 1]};
    v2f bL = {wl[k0], wl[k0 + 1]};
    v2f bR = {wr[k0], wr[k0 + 1]};
    acc = __builtin_amdgcn_wmma_f32_16x16x4_f32(false, aA, false, bL,
                                                (short)0, acc, false, false);
    acc = __builtin_amdgcn_wmma_f32_16x16x4_f32(false, aX, false, bR,
                                                (short)0, acc, false, false);
  }
  // C/D layout: VGPR r = node (base + r + 8*hi), N = wave*16 + (lane&15)
  const float bias  = b1[col];
  const int   nbase = blockIdx.x * 16 + 8 * hi;
#pragma unroll
  for (int r = 0; r < 8; ++r) {
    float v = acc[r] + bias;
    h[(long long)(nbase + r) * HID_CH + col] = fmaxf(v, 0.0f);
  }
}

// ------------------------------------------------------------- layer 2 GEMM
// raw = (agg2/deg) @ W2l.T + h @ W2r.T + b2    [50000,256] -> [50000,40]
// Block: 96 threads = 3 waves (N tiles cover 48 cols; cols >= 40 masked with
// zero B operands so EXEC stays all-ones through the WMMAs).
__global__ __launch_bounds__(96) void gemm2_wmma(
    const float* __restrict__ h, const float* __restrict__ agg2,
    const float* __restrict__ deg, const float* __restrict__ W2l,
    const float* __restrict__ W2r, const float* __restrict__ b2,
    float* __restrict__ raw) {
  const int lane = threadIdx.x & 31;
  const int wave = threadIdx.x >> 5;   // 0..2
  const int hi   = lane >> 4;
  const int m    = lane & 15;
  const int row  = blockIdx.x * 16 + m;
  const int col  = wave * 16 + m;      // 0..47
  const bool colok = (col < OUT_CH);
  const float invd = 1.0f / fmaxf(deg[row], 1.0f);

  const float* hr = h    + (long long)row * HID_CH;
  const float* ar = agg2 + (long long)row * HID_CH;
  const long long wroff = colok ? (long long)col * HID_CH : 0;  // safe addr
  const float* wl = W2l + wroff;
  const float* wr = W2r + wroff;
  const float msk = colok ? 1.0f : 0.0f;

  v8f acc = {};
  for (int kk = 0; kk < HID_CH; kk += 4) {
    const int k0 = kk + 2 * hi;
    v2f aA = {ar[k0] * invd, ar[k0 + 1] * invd};
    v2f aH = {hr[k0], hr[k0 + 1]};
    v2f bL = {wl[k0] * msk, wl[k0 + 1] * msk};
    v2f bR = {wr[k0] * msk, wr[k0 + 1] * msk};
    acc = __builtin_amdgcn_wmma_f32_16x16x4_f32(false, aA, false, bL,
                                                (short)0, acc, false, false);
    acc = __builtin_amdgcn_wmma_f32_16x16x4_f32(false, aH, false, bR,
                                                (short)0, acc, false, false);
  }
  if (colok) {
    const float bias  = b2[col];
    const int   nbase = blockIdx.x * 16 + 8 * hi;
#pragma unroll
    for (int r = 0; r < 8; ++r)
      raw[(long long)(nbase + r) * OUT_CH + col] = acc[r] + bias;
  }
}

// --------------------------------------------------------------- epilogue
__global__ void logsoftmax_kernel(const float* __restrict__ raw,
                                  float* __restrict__ out, int n) {
  int i = blockIdx.x * blockDim.x + threadIdx.x;
  if (i >= n) return;
  const float* r = raw + (long long)i * OUT_CH;
  float mx = r[0];
#pragma unroll
  for (int j = 1; j < OUT_CH; ++j) mx = fmaxf(mx, r[j]);
  float s = 0.0f;
#pragma unroll
  for (int j = 0; j < OUT_CH; ++j) s += __expf(r[j] - mx);
  const float ls = __logf(s);
  float* o = out + (long long)i * OUT_CH;
#pragma unroll
  for (int j = 0; j < OUT_CH; ++j) o[j] = r[j] - mx - ls;
}

// ----------------------------------------------------------------- launcher
extern "C" void kernel_launch(void* const* d_in, const int* in_sizes, int n_in,
                              void* d_out, int out_size, void* d_ws,
                              size_t ws_size, hipStream_t stream) {
  (void)n_in; (void)out_size; (void)ws_size;
  const float* x    = (const float*)d_in[0];
  const int*   eidx = (const int*)d_in[1];   // [2, E] flat, int32
  const float* W1l  = (const float*)d_in[2];
  const float* W1r  = (const float*)d_in[3];
  const float* b1   = (const float*)d_in[4];
  const float* W2l  = (const float*)d_in[5];
  const float* W2r  = (const float*)d_in[6];
  const float* b2   = (const float*)d_in[7];

  const int E = in_sizes[1] / 2;
  const int* src = eidx;
  const int* dst = eidx + E;

  // workspace layout (floats): deg | agg(256-wide, shared) | h | raw
  float* deg = (float*)d_ws;
  float* agg = deg + NNODES;
  float* h   = agg + (size_t)NNODES * HID_CH;
  float* raw = h   + (size_t)NNODES * HID_CH;
  float* out = (float*)d_out;

  const int TB = 256;

  // ---- layer 1 ----
  {  // zero deg + agg (deg and agg are contiguous; layer1 uses 128-wide agg)
    int nz = NNODES * (1 + IN_CH);
    zero_f32<<<(nz + TB - 1) / TB, TB, 0, stream>>>(deg, nz);
  }
  degree_kernel<<<(E + TB - 1) / TB, TB, 0, stream>>>(dst, deg, E);
  {
    int total = E * (IN_CH / 4);
    scatter_add<IN_CH><<<(total + TB - 1) / TB, TB, 0, stream>>>(src, dst, x,
                                                                 agg, E);
  }
  gemm1_wmma<<<NNODES / 16, 512, 0, stream>>>(x, agg, deg, W1l, W1r, b1, h);

  // ---- layer 2 ----
  {
    int nz = NNODES * HID_CH;
    zero_f32<<<(nz + TB - 1) / TB, TB, 0, stream>>>(agg, nz);
  }
  {
    int total = E * (HID_CH / 4);
    scatter_add<HID_CH><<<(total + TB - 1) / TB, TB, 0, stream>>>(src, dst, h,
                                                                  agg, E);
  }
  gemm2_wmma<<<NNODES / 16, 96, 0, stream>>>(h, agg, deg, W2l, W2r, b2, raw);

  // ---- log_softmax ----
  logsoftmax_kernel<<<(NNODES + TB - 1) / TB, TB, 0, stream>>>(raw, out,
                                                               NNODES);
}